// GATLayer_64957085385267
// MI455X (gfx1250) — compile-verified
//
#include <hip/hip_runtime.h>
#include <hip/hip_bf16.h>

// GAT layer for MI455X (gfx1250, wave32).
// Sizes fixed by the reference:
#define NN   100000   // nodes
#define NE   1600000  // edges
#define INF_ 128      // input feats
#define HCF  128      // H*C
#define NH   4        // heads
#define NC   32       // channels/head
#define EDF  16       // edge feats
#define NEG_SLOPE 0.2f
#define BN_EPS 1e-5f

typedef __attribute__((ext_vector_type(2))) float v2f;
typedef __attribute__((ext_vector_type(8))) float v8f;

__device__ __forceinline__ float atomic_add_f32(float* p, float v) {
    return __hip_atomic_fetch_add(p, v, __ATOMIC_RELAXED, __HIP_MEMORY_SCOPE_AGENT);
}

// ---------------------------------------------------------------- init
__global__ void __launch_bounds__(256) k_init(float* __restrict__ out,
                                              int* __restrict__ amax_i,
                                              float* __restrict__ denom,
                                              float* __restrict__ bnsum,
                                              float* __restrict__ bnsumsq) {
    int idx = blockIdx.x * 256 + threadIdx.x;
    if (idx < NN * HCF) out[idx] = 0.0f;
    if (idx < NN * NH) { amax_i[idx] = (int)0x80000000; denom[idx] = 0.0f; }
    if (idx < HCF) { bnsum[idx] = 0.0f; bnsumsq[idx] = 0.0f; }
}

// ---------------------------------------------------------------- h = x @ W
// fp32 WMMA 16x16x4. Block = 16 output rows x 128 cols, 8 waves, wave w owns
// cols [16w,16w+16). x-strip staged in LDS with pitch 132 (conflict-free A reads).
__global__ void __launch_bounds__(256) k_gemm_x(const float* __restrict__ x,
                                                const float* __restrict__ W,
                                                float* __restrict__ h) {
    __shared__ float sx[16 * 132];
    const int tid  = threadIdx.x;
    const int row0 = blockIdx.x * 16;

    // stage 16x128 fp32 strip of x (512 float4), padded pitch 132
    const float4* xg = (const float4*)(x + (size_t)row0 * INF_);
    #pragma unroll
    for (int i = tid; i < 512; i += 256) {
        int r  = i >> 5;          // 32 float4 per row
        int c4 = i & 31;
        ((float4*)(sx + r * 132))[c4] = xg[i];
    }
    __syncthreads();

    const int wave = tid >> 5;
    const int lane = tid & 31;
    const int m    = lane & 15;         // A row / D col index
    const int kb   = (lane >> 4) << 1;  // half-wave K offset (0 or 2)
    const int col0 = wave * 16;

    const float* arow = sx + m * 132 + kb;
    const float* wcol = W + col0 + m;   // B: lane n = lane&15 == m

    v8f c = {};
    #pragma unroll
    for (int k = 0; k < INF_; k += 4) {
        v2f a, b;
        a.x = arow[k];
        a.y = arow[k + 1];
        b.x = wcol[(k + kb) * HCF];
        b.y = wcol[(k + kb + 1) * HCF];
        c = __builtin_amdgcn_wmma_f32_16x16x4_f32(false, a, false, b,
                                                  (short)0, c, false, false);
    }

    // D layout: VGPR j holds M = j + 8*(lane>>4), N = lane&15
    float* hp = h + (size_t)(row0 + 8 * (lane >> 4)) * HCF + col0 + m;
    #pragma unroll
    for (int j = 0; j < 8; ++j) hp[j * HCF] = c[j];
}

// ---------------------------------------------------------------- a_src/a_dst
// one wave32 per node; lane l covers channel l of all 4 heads; shfl_xor reduce.
__global__ void __launch_bounds__(256) k_node_att(const float* __restrict__ h,
                                                  const float* __restrict__ att_src,
                                                  const float* __restrict__ att_dst,
                                                  float* __restrict__ a_src,
                                                  float* __restrict__ a_dst) {
    const int wave = threadIdx.x >> 5, lane = threadIdx.x & 31;
    const int nd = blockIdx.x * 8 + wave;
    if (nd >= NN) return;
    const float* hr = h + (size_t)nd * HCF + lane;
    float ps[NH], pd[NH];
    #pragma unroll
    for (int hh = 0; hh < NH; ++hh) {
        float val = hr[hh * NC];
        ps[hh] = val * att_src[hh * NC + lane];
        pd[hh] = val * att_dst[hh * NC + lane];
    }
    #pragma unroll
    for (int off = 16; off > 0; off >>= 1) {
        #pragma unroll
        for (int hh = 0; hh < NH; ++hh) {
            ps[hh] += __shfl_xor(ps[hh], off, 32);
            pd[hh] += __shfl_xor(pd[hh], off, 32);
        }
    }
    if (lane == 0) {
        #pragma unroll
        for (int hh = 0; hh < NH; ++hh) {
            a_src[nd * NH + hh] = ps[hh];
            a_dst[nd * NH + hh] = pd[hh];
        }
    }
}

// ---------------------------------------------------------------- v = fold(W_edge, att_edge)
// v[h][d] = sum_c W_edge[d, 32h+c] * att_edge[h,c]  (eliminates the [E,128] GEMM)
__global__ void k_vedge(const float* __restrict__ W_edge,
                        const float* __restrict__ att_edge,
                        float* __restrict__ v) {
    int t = threadIdx.x;
    if (t >= NH * EDF) return;
    int hh = t >> 4, d = t & 15;
    float s = 0.0f;
    #pragma unroll
    for (int cc = 0; cc < NC; ++cc)
        s += W_edge[d * HCF + hh * NC + cc] * att_edge[hh * NC + cc];
    v[t] = s;
}

// ---------------------------------------------------------------- alpha + segment max
__device__ __forceinline__ int f32_key(float f) {
    int b = __float_as_int(f);
    return b >= 0 ? b : (b ^ 0x7fffffff);  // monotone float -> signed-int key
}

__global__ void __launch_bounds__(256) k_alpha(const int* __restrict__ ei,
                                               const float* __restrict__ ea,
                                               const float* __restrict__ v,
                                               const float* __restrict__ a_src,
                                               const float* __restrict__ a_dst,
                                               float* __restrict__ alpha,
                                               int* __restrict__ amax_i) {
    int e = blockIdx.x * 256 + threadIdx.x;
    if (e >= NE) return;
    const int s = ei[e], d = ei[NE + e];
    float ef[EDF];
    const float* er = ea + (size_t)e * EDF;
    #pragma unroll
    for (int i = 0; i < EDF; ++i) ef[i] = er[i];
    #pragma unroll
    for (int hh = 0; hh < NH; ++hh) {
        float ae = 0.0f;
        #pragma unroll
        for (int i = 0; i < EDF; ++i) ae += ef[i] * v[hh * EDF + i];
        float al = a_src[s * NH + hh] + a_dst[d * NH + hh] + ae;
        al = al > 0.0f ? al : NEG_SLOPE * al;         // leaky relu
        alpha[e * NH + hh] = al;
        atomicMax(&amax_i[d * NH + hh], f32_key(al)); // global_atomic_max_i32
    }
}

// ---------------------------------------------------------------- decode amax (in place)
__global__ void __launch_bounds__(256) k_amax_fix(int* __restrict__ amax_i) {
    int idx = blockIdx.x * 256 + threadIdx.x;
    if (idx >= NN * NH) return;
    int k = amax_i[idx];
    float f = 0.0f;
    if (k != (int)0x80000000) {
        int b = k >= 0 ? k : (k ^ 0x7fffffff);
        f = __int_as_float(b);
        if (!(f > -3.0e38f && f < 3.0e38f)) f = 0.0f;  // where(isfinite, amax, 0)
    }
    ((float*)amax_i)[idx] = f;
}

// ---------------------------------------------------------------- ex = exp(alpha-amax), denom += ex
__global__ void __launch_bounds__(256) k_exp(const int* __restrict__ ei,
                                             const float* __restrict__ amaxf,
                                             float* __restrict__ alpha,
                                             float* __restrict__ denom) {
    int e = blockIdx.x * 256 + threadIdx.x;
    if (e >= NE) return;
    const int d = ei[NE + e];
    #pragma unroll
    for (int hh = 0; hh < NH; ++hh) {
        float ex = __expf(alpha[e * NH + hh] - amaxf[d * NH + hh]);
        alpha[e * NH + hh] = ex;                        // reuse buffer for ex
        atomic_add_f32(&denom[d * NH + hh], ex);        // global_atomic_add_f32
    }
}

// ---------------------------------------------------------------- weighted scatter (hot loop)
// one wave32 per edge; lane l handles channel l of each head -> 4 loads + 4 f32 atomics.
// h (51MB) and out (51MB) both fit in the 192MB L2 -> L2-resident gather/scatter.
__global__ void __launch_bounds__(256) k_scatter(const int* __restrict__ ei,
                                                 const float* __restrict__ ex,
                                                 const float* __restrict__ denom,
                                                 const float* __restrict__ h,
                                                 float* __restrict__ out) {
    const int wave = threadIdx.x >> 5, lane = threadIdx.x & 31;
    const int e = blockIdx.x * 8 + wave;
    if (e >= NE) return;
    const int s = ei[e], d = ei[NE + e];
    float w[NH];
    #pragma unroll
    for (int hh = 0; hh < NH; ++hh)
        w[hh] = ex[e * NH + hh] / (denom[d * NH + hh] + 1e-16f);
    const float* hr = h + (size_t)s * HCF + lane;
    float* orow = out + (size_t)d * HCF + lane;
    #pragma unroll
    for (int hh = 0; hh < NH; ++hh)
        atomic_add_f32(&orow[hh * NC], w[hh] * hr[hh * NC]);
}

// ---------------------------------------------------------------- BN stats (bias cancels in BN)
__global__ void __launch_bounds__(256) k_bnstats(const float* __restrict__ out,
                                                 float* __restrict__ bnsum,
                                                 float* __restrict__ bnsumsq) {
    const int c = threadIdx.x & 127;
    const int rg = threadIdx.x >> 7;  // two row streams per block
    float s = 0.0f, sq = 0.0f;
    for (int n = blockIdx.x * 2 + rg; n < NN; n += gridDim.x * 2) {
        float val = out[(size_t)n * HCF + c];
        s += val; sq += val * val;
    }
    atomic_add_f32(&bnsum[c], s);
    atomic_add_f32(&bnsumsq[c], sq);
}

// ---------------------------------------------------------------- normalize + relu (in place)
__global__ void __launch_bounds__(256) k_norm(float* __restrict__ out,
                                              const float* __restrict__ bnsum,
                                              const float* __restrict__ bnsumsq,
                                              const float* __restrict__ gamma,
                                              const float* __restrict__ beta) {
    int idx = blockIdx.x * 256 + threadIdx.x;
    if (idx >= NN * HCF) return;
    int c = idx & 127;
    const float invN = 1.0f / (float)NN;
    float mean = bnsum[c] * invN;
    float var  = bnsumsq[c] * invN - mean * mean;
    float y = (out[idx] - mean) * rsqrtf(var + BN_EPS) * gamma[c] + beta[c];
    out[idx] = y > 0.0f ? y : 0.0f;
}

// ---------------------------------------------------------------- launch
extern "C" void kernel_launch(void* const* d_in, const int* in_sizes, int n_in,
                              void* d_out, int out_size, void* d_ws, size_t ws_size,
                              hipStream_t stream) {
    const float* x        = (const float*)d_in[0];
    const int*   ei       = (const int*)  d_in[1];
    const float* ea       = (const float*)d_in[2];
    const float* W        = (const float*)d_in[3];
    const float* W_edge   = (const float*)d_in[4];
    const float* att_src  = (const float*)d_in[5];
    const float* att_dst  = (const float*)d_in[6];
    const float* att_edge = (const float*)d_in[7];
    // d_in[8] = bias: provably cancels inside batch-norm (constant shift of mean)
    const float* gamma    = (const float*)d_in[9];
    const float* beta     = (const float*)d_in[10];
    float* out = (float*)d_out;

    // workspace layout (float elements), total ~83.2 MB
    float* ws      = (float*)d_ws;
    float* h       = ws;                  // NN*128   = 12,800,000
    float* a_src   = ws + 12800000;       // NN*4
    float* a_dst   = ws + 13200000;       // NN*4
    float* alpha   = ws + 13600000;       // NE*4     =  6,400,000 (reused as ex)
    int*   amax_i  = (int*)(ws + 20000000); // NN*4 (int keys, decoded to float in place)
    float* amaxf   = (float*)amax_i;
    float* denom   = ws + 20400000;       // NN*4
    float* v       = ws + 20800000;       // 64
    float* bnsum   = ws + 20800128;       // 128
    float* bnsumsq = ws + 20800256;       // 128

    k_init    <<<50000, 256, 0, stream>>>(out, amax_i, denom, bnsum, bnsumsq);
    k_gemm_x  <<<NN / 16, 256, 0, stream>>>(x, W, h);                 // 6250 blocks
    k_node_att<<<NN / 8, 256, 0, stream>>>(h, att_src, att_dst, a_src, a_dst);
    k_vedge   <<<1, 64, 0, stream>>>(W_edge, att_edge, v);
    k_alpha   <<<NE / 256, 256, 0, stream>>>(ei, ea, v, a_src, a_dst, alpha, amax_i);
    k_amax_fix<<<(NN * NH + 255) / 256, 256, 0, stream>>>(amax_i);
    k_exp     <<<NE / 256, 256, 0, stream>>>(ei, amaxf, alpha, denom);
    k_scatter <<<NE / 8, 256, 0, stream>>>(ei, alpha, denom, h, out); // 200000 blocks
    k_bnstats <<<256, 256, 0, stream>>>(out, bnsum, bnsumsq);
    k_norm    <<<50000, 256, 0, stream>>>(out, bnsum, bnsumsq, gamma, beta);
}